// VectorQuantizer_64467459113392
// MI455X (gfx1250) — compile-verified
//
#include <hip/hip_runtime.h>

// VQ-VAE vector quantizer for MI455X (gfx1250).
// N = 32768 pixels, C = 256 dims, K = 1024 codes.
#define N_ROWS   32768
#define C_DIM    256
#define K_CODES  1024
#define LDSTRIDE 260   // 256 + 4 pad: LDS bank = (4*row + k) % 64 -> conflict-free frag reads

typedef __attribute__((ext_vector_type(2))) float v2f;
typedef __attribute__((ext_vector_type(8))) float v8f;

// ---------------- Kernel 1: codebook squared norms ----------------
__global__ void __launch_bounds__(128)
vq_cnorm_kernel(const float* __restrict__ cb, float* __restrict__ cnorm) {
  int code = blockIdx.x * 128 + threadIdx.x;          // grid 8 x 128 = 1024
  const float4* row = (const float4*)(cb + (size_t)code * C_DIM);
  float s = 0.f;
#pragma unroll 8
  for (int i = 0; i < C_DIM / 4; ++i) {
    float4 v = row[i];
    s = fmaf(v.x, v.x, s); s = fmaf(v.y, v.y, s);
    s = fmaf(v.z, v.z, s); s = fmaf(v.w, v.w, s);
  }
  cnorm[code] = s;
}

// Issue one 32-code x 256-dim tile (32 KB) as per-lane async B128 copies into LDS.
// Each of the 128 threads issues 16 global_load_async_to_lds_b128 (ASYNCcnt-tracked).
__device__ __forceinline__ void stage_tile_async(const float* __restrict__ cb,
                                                 int tile, float* lds_dst, int tid) {
  const uint32_t ldsbase = (uint32_t)(uintptr_t)(void*)lds_dst;  // low 32b = LDS offset
#pragma unroll
  for (int j = 0; j < 16; ++j) {
    int i    = tid + j * 128;          // 2048 float4 tasks
    int code = i >> 6;                 // 0..31
    int q    = i & 63;                 // float4 index within row
    uint64_t ga = (uint64_t)(uintptr_t)(cb + ((size_t)(tile * 32 + code) * C_DIM + 4 * q));
    uint32_t la = ldsbase + (uint32_t)((code * LDSTRIDE + 4 * q) * sizeof(float));
    asm volatile("global_load_async_to_lds_b128 %0, %1, off"
                 :: "v"(la), "v"(ga) : "memory");
  }
}

__device__ __forceinline__ void wait_async_all() {
  asm volatile("s_wait_asynccnt 0x0" ::: "memory");
}

// ---------------- Kernel 2: distances (f32 WMMA) + argmin ----------------
// block = 128 threads (4 waves), each block owns 64 consecutive pixels (one
// contiguous hw strip within one image), each wave owns 16 rows.
// argmin metric: ||c||^2 - 2 z.c   (||z||^2 is constant per row).
// Codebook tiles double-buffered in LDS via async copies overlapped with WMMA.
__global__ void __launch_bounds__(128)
vq_argmin_kernel(const float* __restrict__ z, const float* __restrict__ cb,
                 const float* __restrict__ cnorm, int* __restrict__ idx_out) {
  extern __shared__ float smem[];
  float* zt   = smem;                                   // 64 rows x LDSTRIDE
  float* cbuf0 = smem + 64 * LDSTRIDE;                  // 32 codes x LDSTRIDE
  float* cbuf1 = cbuf0 + 32 * LDSTRIDE;                 // 32 codes x LDSTRIDE

  const int tid  = threadIdx.x;
  const int wave = tid >> 5;
  const int lane = tid & 31;
  const int lm   = lane & 15;   // row/col within 16
  const int lh   = lane >> 4;   // half-wave select (K pair / M+8)

  const int rowbase = blockIdx.x * 64;       // pixel row base
  const int b       = rowbase >> 10;         // image index
  const int hw0     = rowbase & 1023;        // contiguous hw start
  const size_t zbase = (size_t)b * (C_DIM * 1024) + hw0;

  // Kick off async staging of codebook tile 0 into buffer 0 (overlaps z staging).
  stage_tile_async(cb, 0, cbuf0, tid);

  // Stage z tile transposed: global read coalesced along hw, LDS as [row][c].
  for (int i = tid; i < 8192; i += 128) {    // 256 c * 32 float2 strips
    int c = i >> 5;
    int p = i & 31;
    float2 v = *(const float2*)(z + zbase + (size_t)c * 1024 + 2 * p);
    zt[(2 * p    ) * LDSTRIDE + c] = v.x;
    zt[(2 * p + 1) * LDSTRIDE + c] = v.y;
  }

  float bd[8]; int bi[8];
#pragma unroll
  for (int v = 0; v < 8; ++v) { bd[v] = 3.4e38f; bi[v] = 0; }

  // Fragment base offsets: lane reads 2 consecutive floats at [l%16][2*(l/16)+4*ks]
  const float* Ab = zt + (wave * 16 + lm) * LDSTRIDE + 2 * lh;
  const int boff0 = lm * LDSTRIDE + 2 * lh;
  const int boff1 = (16 + lm) * LDSTRIDE + 2 * lh;

  wait_async_all();          // tile 0 resident
  __syncthreads();

  for (int tt = 0; tt < 32; ++tt) {          // 32 iterations x 32 codes
    float* cur = (tt & 1) ? cbuf1 : cbuf0;
    if (tt + 1 < 32)                         // prefetch next tile into other buffer
      stage_tile_async(cb, tt + 1, (tt & 1) ? cbuf0 : cbuf1, tid);

    const float* B0b = cur + boff0;
    const float* B1b = cur + boff1;

    v8f acc0 = {}; v8f acc1 = {};
#pragma unroll 4
    for (int ks = 0; ks < 64; ++ks) {        // K = 256 in steps of 4
      v2f a  = *(const v2f*)(Ab  + ks * 4);
      v2f b0 = *(const v2f*)(B0b + ks * 4);
      v2f b1 = *(const v2f*)(B1b + ks * 4);
      acc0 = __builtin_amdgcn_wmma_f32_16x16x4_f32(false, a, false, b0,
                                                   (short)0, acc0, false, false);
      acc1 = __builtin_amdgcn_wmma_f32_16x16x4_f32(false, a, false, b1,
                                                   (short)0, acc1, false, false);
    }

    const float cn0 = cnorm[tt * 32 + lm];        // L2-resident
    const float cn1 = cnorm[tt * 32 + 16 + lm];
    const int code0 = tt * 32 + lm;
    const int code1 = code0 + 16;
#pragma unroll
    for (int v = 0; v < 8; ++v) {
      float d0 = fmaf(-2.f, acc0[v], cn0);
      float d1 = fmaf(-2.f, acc1[v], cn1);
      if (d0 < bd[v]) { bd[v] = d0; bi[v] = code0; }   // strict <: keep lowest idx
      if (d1 < bd[v]) { bd[v] = d1; bi[v] = code1; }
    }

    wait_async_all();        // next tile resident; also fences before buffer reuse
    __syncthreads();
  }

  // Cross-lane argmin within each 16-lane half (codes live across lanes).
#pragma unroll
  for (int v = 0; v < 8; ++v) {
    float d = bd[v]; int i = bi[v];
#pragma unroll
    for (int m = 8; m >= 1; m >>= 1) {
      float od = __shfl_xor(d, m, 32);
      int   oi = __shfl_xor(i, m, 32);
      if (od < d || (od == d && oi < i)) { d = od; i = oi; }
    }
    if (lm == 0) {                       // lane 0 -> row v, lane 16 -> row v+8
      int row = rowbase + wave * 16 + v + (lh ? 8 : 0);
      idx_out[row] = i;
    }
  }
}

// ---------------- Kernel 3: gather z_q, straight-through output, loss partials ----
__global__ void __launch_bounds__(256)
vq_gather_kernel(const float* __restrict__ z, const float* __restrict__ cb,
                 const int* __restrict__ idx, float* __restrict__ out,
                 float* __restrict__ partial) {
  __shared__ float red[256];
  float sum = 0.f;
  const size_t base = (size_t)blockIdx.x * 2048 + threadIdx.x;  // 4096 blocks
#pragma unroll
  for (int j = 0; j < 8; ++j) {
    size_t e = base + (size_t)j * 256;        // NCHW flat index, coalesced
    int hw = (int)(e & 1023);
    int c  = (int)((e >> 10) & 255);
    int b  = (int)(e >> 18);
    int n  = (b << 10) | hw;                  // pixel row
    float zv = z[e];
    float q  = cb[(size_t)idx[n] * C_DIM + c];
    float t  = q - zv;                        // stop_gradient(z_q - z)
    out[e]   = zv + t;                        // straight-through estimator
    sum = fmaf(t, t, sum);
  }
  red[threadIdx.x] = sum;
  __syncthreads();
  for (int s = 128; s > 0; s >>= 1) {
    if (threadIdx.x < s) red[threadIdx.x] += red[threadIdx.x + s];
    __syncthreads();
  }
  if (threadIdx.x == 0) partial[blockIdx.x] = red[0];
}

// ---------------- Kernel 4: deterministic final loss reduction ----------------
__global__ void __launch_bounds__(256)
vq_loss_kernel(const float* __restrict__ partial, float* __restrict__ loss_out) {
  __shared__ float red[256];
  float s = 0.f;
  for (int i = threadIdx.x; i < 4096; i += 256) s += partial[i];
  red[threadIdx.x] = s;
  __syncthreads();
  for (int k = 128; k > 0; k >>= 1) {
    if (threadIdx.x < k) red[threadIdx.x] += red[threadIdx.x + k];
    __syncthreads();
  }
  if (threadIdx.x == 0) {
    float m = red[0] * (1.0f / 8388608.0f);   // mean over 32*256*32*32
    loss_out[0] = m + 0.25f * m;              // mse + BETA * mse
  }
}

// ---------------- launch ----------------
extern "C" void kernel_launch(void* const* d_in, const int* in_sizes, int n_in,
                              void* d_out, int out_size, void* d_ws, size_t ws_size,
                              hipStream_t stream) {
  const float* z  = (const float*)d_in[0];   // (32,256,32,32) f32
  const float* cb = (const float*)d_in[1];   // (1024,256) f32
  float* out = (float*)d_out;                // 8388608 z_q_st + 1 loss

  float* cnorm   = (float*)d_ws;                              // 1024 f32
  int*   idx     = (int*)  ((char*)d_ws + 4096);              // 32768 i32
  float* partial = (float*)((char*)d_ws + 4096 + 131072);     // 4096 f32

  vq_cnorm_kernel<<<8, 128, 0, stream>>>(cb, cnorm);

  size_t shmem = (size_t)(64 * LDSTRIDE + 2 * 32 * LDSTRIDE) * sizeof(float);
  vq_argmin_kernel<<<N_ROWS / 64, 128, shmem, stream>>>(z, cb, cnorm, idx);

  vq_gather_kernel<<<4096, 256, 0, stream>>>(z, cb, idx, out, partial);
  vq_loss_kernel<<<1, 256, 0, stream>>>(partial, out + 8388608);
}